// VectorQuantizerEMA_35141422415995
// MI455X (gfx1250) — compile-verified
//
#include <hip/hip_runtime.h>
#include <hip/hip_bf16.h>
#include <math.h>
#include <stdint.h>

typedef __attribute__((ext_vector_type(2))) float v2f;
typedef __attribute__((ext_vector_type(8))) float v8f;
typedef __attribute__((ext_vector_type(4))) unsigned int u32x4;
typedef __attribute__((ext_vector_type(8))) int i32x8;
typedef __attribute__((ext_vector_type(4))) int i32x4;

// Problem geometry
constexpr int ED      = 256;     // embedding dim (GEMM K-depth)
constexpr int NCODE   = 8192;    // codebook size (GEMM N)
constexpr int NTOKENS = 8192;    // 8 * 32 * 32   (GEMM M)
constexpr int HWSZ    = 1024;    // 32*32
constexpr int NB      = 8;

// d_out layout (tuple concatenated flat, all float32):
// quantized_st [2097152] | loss [1] | perplexity [1] | idx [8192] | dw [2097152] | cluster [8192]
constexpr size_t Q_OFF    = 0;
constexpr size_t LOSS_OFF = 2097152;
constexpr size_t PERP_OFF = 2097153;
constexpr size_t IDX_OFF  = 2097154;
constexpr size_t DW_OFF   = 2105346;
constexpr size_t CS_OFF   = 4202498;   // == DW_OFF + ED*NCODE (contiguous with dw)

// ws layout (floats): [0]=sse accum, [64..64+8192)=enorm, then 8192 ints of winner idx
constexpr int WS_ENORM_OFF = 64;
constexpr int WS_IDX_OFF   = 64 + NCODE;

// ---------------- zero accumulators (dw + cluster_size + sse) ----------------
__global__ void k_zero(float* __restrict__ out, float* __restrict__ sse) {
    int i = blockIdx.x * 256 + threadIdx.x;
    int n = ED * NCODE + NCODE;                 // dw + cluster, contiguous
    if (i < n) out[DW_OFF + i] = 0.0f;
    if (i == 0) *sse = 0.0f;
}

// ---------------- per-code squared norms |e_k|^2 ----------------
__global__ void k_enorm(const float* __restrict__ emb, float* __restrict__ enorm) {
    int k = blockIdx.x * 256 + threadIdx.x;     // coalesced across k
    float s = 0.0f;
    for (int d = 0; d < ED; ++d) { float v = emb[(size_t)d * NCODE + k]; s += v * v; }
    enorm[k] = s;
}

// ---- TDM: one 2D D# describing a 256x16 f32 B-tile of the codebook -> LDS ----
// Tile columns kt*16..kt*16+15 of emb[256][8192] land contiguously as Bt[d*16+n].
__device__ inline void tdm_load_b_tile(const float* emb, int kt, unsigned lds_off) {
    unsigned long long ga = (unsigned long long)(uintptr_t)(emb + (size_t)kt * 16);
    u32x4 g0;
    g0[0] = 1u;                                          // count=1 user descriptor
    g0[1] = lds_off;                                     // lds_addr (bytes)
    g0[2] = (unsigned)(ga & 0xffffffffu);                // global_addr[31:0]
    g0[3] = (unsigned)((ga >> 32) & 0x01ffffffu)         // global_addr[56:32]
          | (2u << 30);                                  // type = 2 ("image")
    i32x8 g1;
    g1[0] = (int)(2u << 16);        // workgroup_mask=0 | data_size=2 (4 bytes)
    g1[1] = (int)(8192u << 16);     // atomic_barrier_addr=0 | tensor_dim0[15:0]=8192
    g1[2] = (int)(256u << 16);      // tensor_dim0[31:16]=0 | tensor_dim1[15:0]=256
    g1[3] = (int)(16u << 16);       // tensor_dim1[31:16]=0 | tile_dim0=16
    g1[4] = (int)(256u);            // tile_dim1=256 | tile_dim2=0
    g1[5] = (int)(8192u);           // tensor_dim0_stride[31:0]=8192
    g1[6] = 0;                      // stride0 hi | stride1 lo (unused for 2D)
    g1[7] = 0;
    i32x4 gz4 = {0, 0, 0, 0};       // groups 2/3 zero: 2D tensor
    i32x8 gz8 = {0, 0, 0, 0, 0, 0, 0, 0};
    __builtin_amdgcn_tensor_load_to_lds(g0, g1, gz4, gz4, gz8, 0);
}

// ---------------- fused distance-GEMM + argmin (WMMA f32 + TDM) ----------------
// 4 waves/WG, 16 tokens/wave => 64 tokens/WG, 128 WGs.
// A-tile (16x256 f32) resident in registers per wave; B tiles double-buffered in
// LDS via Tensor Data Mover (wave 0 issues, TENSORcnt + one barrier per tile).
__global__ __launch_bounds__(128)
void k_argmin(const float* __restrict__ z, const float* __restrict__ emb,
              const float* __restrict__ enorm_g,
              float* __restrict__ out, int* __restrict__ idxbuf) {
    __shared__ float eLDS[NCODE];        // 32 KB: |e|^2 resident in LDS
    __shared__ float Bt[2][16 * ED];     // 2 x 16 KB: double-buffered B tiles
    __shared__ float redD[4 * 16 * 16];
    __shared__ int   redI[4 * 16 * 16];

    const int tid  = threadIdx.x;
    const int lane = tid & 31;
    const int wv   = tid >> 5;
    const int ln   = lane & 15;
    const int hi   = lane >> 4;          // half-wave: selects K pair per WMMA layout

    const unsigned ldsB0 = (unsigned)(uintptr_t)&Bt[0][0];   // LDS byte offsets
    const unsigned ldsB1 = (unsigned)(uintptr_t)&Bt[1][0];

    // kick off first B tile via TDM (wave 0 only; uniform scalar branch)
    if (wv == 0) tdm_load_b_tile(emb, 0, ldsB0);

    // stage |e|^2 into LDS (float4, coalesced) while the DMA runs
    {
        const float4* src = (const float4*)enorm_g;
        float4* dst = (float4*)eLDS;
        for (int i = tid; i < NCODE / 4; i += 128) dst[i] = src[i];
    }

    // A fragments for V_WMMA_F32_16X16X4_F32:
    // lanes 0-15 hold (M=lane, K=4f..4f+1), lanes 16-31 hold (M=lane-16, K=4f+2..4f+3).
    // Token M maps to z NCHW directly: consecutive tokens are consecutive hw -> coalesced.
    const int m0  = blockIdx.x * 64 + wv * 16;
    const int tok = m0 + ln;
    const int bb  = tok >> 10;
    const int hw  = tok & 1023;
    const float* zbase = z + (size_t)bb * (ED * HWSZ) + hw;   // d-stride = HWSZ
    v2f afrag[64];
    #pragma unroll
    for (int f = 0; f < 64; ++f) {
        int kd = 4 * f + 2 * hi;
        afrag[f].x = zbase[(size_t)kd * HWSZ];
        afrag[f].y = zbase[(size_t)(kd + 1) * HWSZ];
    }

    float bestd[8];
    int   besti[8];
    #pragma unroll
    for (int i = 0; i < 8; ++i) { bestd[i] = 3.0e38f; besti[i] = 0; }

    int cur = 0;
    for (int kt = 0; kt < NCODE / 16; ++kt) {
        // wave 0 owns TENSORcnt; barrier publishes the tile (and retires the
        // other buffer's consumers from the previous iteration)
        if (wv == 0) __builtin_amdgcn_s_wait_tensorcnt(0);
        __syncthreads();
        if (wv == 0 && kt + 1 < NCODE / 16)
            tdm_load_b_tile(emb, kt + 1, cur ? ldsB0 : ldsB1);

        const float* Btc = &Bt[cur][0];
        auto ldb = [&](int kd) -> v2f {
            v2f bf;
            bf.x = Btc[kd * 16 + ln];
            bf.y = Btc[(kd + 1) * 16 + ln];
            return bf;
        };

        // 64 WMMAs over depth, 4 independent accumulator chains for ILP
        v8f c0 = {0,0,0,0,0,0,0,0}, c1 = {0,0,0,0,0,0,0,0};
        v8f c2 = {0,0,0,0,0,0,0,0}, c3 = {0,0,0,0,0,0,0,0};
        #pragma unroll
        for (int f = 0; f < 64; f += 4) {
            int h2 = 2 * hi;
            c0 = __builtin_amdgcn_wmma_f32_16x16x4_f32(false, afrag[f+0], false, ldb(4*(f+0)+h2), (short)0, c0, false, false);
            c1 = __builtin_amdgcn_wmma_f32_16x16x4_f32(false, afrag[f+1], false, ldb(4*(f+1)+h2), (short)0, c1, false, false);
            c2 = __builtin_amdgcn_wmma_f32_16x16x4_f32(false, afrag[f+2], false, ldb(4*(f+2)+h2), (short)0, c2, false, false);
            c3 = __builtin_amdgcn_wmma_f32_16x16x4_f32(false, afrag[f+3], false, ldb(4*(f+3)+h2), (short)0, c3, false, false);
        }

        // distance criterion: |e|^2 - 2*dot  (|x|^2 constant per token, drop it)
        int ng = kt * 16 + ln;
        float en = eLDS[ng];
        #pragma unroll
        for (int i = 0; i < 8; ++i) {
            float dot  = c0[i] + c1[i] + c2[i] + c3[i];
            float dist = en - 2.0f * dot;
            if (dist < bestd[i]) { bestd[i] = dist; besti[i] = ng; }   // strict < => lowest idx on tie
        }
        cur ^= 1;
    }

    // cross-lane argmin reduction via LDS (C/D layout: vgpr i, half hi -> M = i + 8*hi)
    #pragma unroll
    for (int i = 0; i < 8; ++i) {
        int m = i + 8 * hi;
        redD[wv * 256 + m * 16 + ln] = bestd[i];
        redI[wv * 256 + m * 16 + ln] = besti[i];
    }
    __syncthreads();
    if (tid < 64) {
        int w = tid >> 4, m = tid & 15;
        float bd = redD[w * 256 + m * 16];
        int   bi = redI[w * 256 + m * 16];
        #pragma unroll
        for (int l = 1; l < 16; ++l) {
            float dd2 = redD[w * 256 + m * 16 + l];
            int   ii  = redI[w * 256 + m * 16 + l];
            if (dd2 < bd || (dd2 == bd && ii < bi)) { bd = dd2; bi = ii; }
        }
        int ntok = blockIdx.x * 64 + tid;
        idxbuf[ntok] = bi;
        out[IDX_OFF + ntok] = (float)bi;
        atomicAdd(&out[CS_OFF + bi], 1.0f);
    }
}

// ---------------- quantize gather + dw scatter + SSE ----------------
// One block per (b,d) row: coalesced z reads and quantized writes; codebook
// row d (32 KB) gathers hit L2; dw atomics scatter within row d of dw.
__global__ __launch_bounds__(256)
void k_quant(const float* __restrict__ z, const float* __restrict__ emb,
             const int* __restrict__ idxbuf, float* __restrict__ out,
             float* __restrict__ sse) {
    __shared__ float red[256];
    int bid = blockIdx.x;
    int b = bid >> 8;
    int d = bid & 255;
    int t = threadIdx.x;
    size_t base = (size_t)b * (ED * HWSZ) + (size_t)d * HWSZ;
    int hw0 = t * 4;
    float4 zv = *(const float4*)(z + base + hw0);
    float4 qv;
    float* qp = (float*)&qv;
    const float* zp = (const float*)&zv;
    float acc = 0.0f;
    #pragma unroll
    for (int j = 0; j < 4; ++j) {
        int idx = idxbuf[b * HWSZ + hw0 + j];
        float q = emb[(size_t)d * NCODE + idx];
        qp[j] = q;
        float df = q - zp[j];
        acc += df * df;
        atomicAdd(&out[DW_OFF + (size_t)d * NCODE + idx], zp[j]);
    }
    *(float4*)(out + Q_OFF + base + hw0) = qv;     // straight-through fwd value == q
    red[t] = acc;
    __syncthreads();
    for (int s = 128; s > 0; s >>= 1) {
        if (t < s) red[t] += red[t + s];
        __syncthreads();
    }
    if (t == 0) atomicAdd(sse, red[0]);
}

// ---------------- scalars: loss + perplexity ----------------
__global__ void k_final(const float* __restrict__ sse, float* __restrict__ out) {
    __shared__ float red[256];
    int t = threadIdx.x;
    float acc = 0.0f;
    for (int k = t; k < NCODE; k += 256) {
        float p = out[CS_OFF + k] * (1.0f / (float)NTOKENS);
        acc += p * logf(p + 1e-10f);
    }
    red[t] = acc;
    __syncthreads();
    for (int s = 128; s > 0; s >>= 1) {
        if (t < s) red[t] += red[t + s];
        __syncthreads();
    }
    if (t == 0) {
        out[PERP_OFF] = expf(-red[0]);
        out[LOSS_OFF] = 0.25f * (*sse) / (float)(NB * ED * HWSZ);
    }
}

extern "C" void kernel_launch(void* const* d_in, const int* in_sizes, int n_in,
                              void* d_out, int out_size, void* d_ws, size_t ws_size,
                              hipStream_t stream) {
    (void)in_sizes; (void)n_in; (void)out_size; (void)ws_size;
    const float* z   = (const float*)d_in[0];
    const float* emb = (const float*)d_in[1];
    float* out = (float*)d_out;
    float* ws  = (float*)d_ws;
    float* sse   = ws;
    float* enorm = ws + WS_ENORM_OFF;
    int*   idxb  = (int*)(ws + WS_IDX_OFF);

    k_zero  <<<(ED * NCODE + NCODE + 255) / 256, 256, 0, stream>>>(out, sse);
    k_enorm <<<NCODE / 256, 256, 0, stream>>>(emb, enorm);
    k_argmin<<<NTOKENS / 64, 128, 0, stream>>>(z, emb, enorm, out, idxb);
    k_quant <<<NB * ED, 256, 0, stream>>>(z, emb, idxb, out, sse);
    k_final <<<1, 256, 0, stream>>>(sse, out);
}